// GumbelIndexer_42674795053914
// MI455X (gfx1250) — compile-verified
//
#include <hip/hip_runtime.h>

// ---------------------------------------------------------------------------
// GumbelIndexer for MI455X (gfx1250, wave32, WMMA + async-to-LDS pipeline)
//   logits = x @ w^T  via v_wmma_f32_16x16x4_f32, fused gumbel+argmax
//   (logits never hit memory), double-buffered LDS staged with
//   global_load_async_to_lds_b128 (ASYNCcnt), streaming traffic marked NT
//   so x/w/embeddings stay L2-resident (75 MB << 192 MB L2).
// ---------------------------------------------------------------------------

typedef float v2f __attribute__((ext_vector_type(2)));
typedef float v4f __attribute__((ext_vector_type(4)));
typedef float v8f __attribute__((ext_vector_type(8)));

#define B_   4
#define S_   2048
#define D_   768
#define K_   8192
#define NS_  2
#define M_   (B_ * S_)          // 8192 rows

#define BM   64                 // rows per workgroup
#define BN   128                // codes per n-block
#define BK   32                 // k-chunk through LDS
#define LDA  36                 // padded LDS row stride (floats)
#define KCH  (D_ / BK)          // 24 k-chunks

__global__ __launch_bounds__(256)
void gumbel_gemm_argmax_kernel(const float* __restrict__ x,
                               const float* __restrict__ w,
                               const float* __restrict__ gn,
                               int* __restrict__ idx_out) {
    __shared__ float Asub[2][BM][LDA];
    __shared__ float Bsub[2][BN][LDA];

    const int tid  = threadIdx.x;
    const int lane = tid & 31;
    const int wv   = tid >> 5;          // wave 0..7
    const int wm   = wv & 3;            // m sub-tile (16 rows each)
    const int wn   = wv >> 2;           // n half (64 cols each)
    const int row0 = blockIdx.x * BM;

    const int l16 = lane & 15;
    const int lhi = lane >> 4;          // 0 or 1

    // async stage A chunk: BM x BK floats = 512 float4, 2 per thread
    auto stageA = [&](int buf, int kc) {
#pragma unroll
        for (int it = 0; it < (BM * BK / 4) / 256; ++it) {
            const int e  = tid + it * 256;
            const int rr = e >> 3;
            const int c4 = e & 7;
            const float*   g = &x[(size_t)(row0 + rr) * D_ + kc + c4 * 4];
            const uint32_t l = (uint32_t)(uintptr_t)&Asub[buf][rr][c4 * 4];
            asm volatile("global_load_async_to_lds_b128 %0, %1, off"
                         :: "v"(l), "v"(g) : "memory");
        }
    };
    // async stage B chunk: BN x BK floats = 1024 float4, 4 per thread
    auto stageB = [&](int buf, int n0, int kc) {
#pragma unroll
        for (int it = 0; it < (BN * BK / 4) / 256; ++it) {
            const int e  = tid + it * 256;
            const int rr = e >> 3;
            const int c4 = e & 7;
            const float*   g = &w[(size_t)(n0 + rr) * D_ + kc + c4 * 4];
            const uint32_t l = (uint32_t)(uintptr_t)&Bsub[buf][rr][c4 * 4];
            asm volatile("global_load_async_to_lds_b128 %0, %1, off"
                         :: "v"(l), "v"(g) : "memory");
        }
    };

    // running best (value, index) per sample per C-fragment VGPR slot
    float bestv[NS_][8];
    int   besti[NS_][8];
#pragma unroll
    for (int s = 0; s < NS_; ++s)
#pragma unroll
        for (int v = 0; v < 8; ++v) { bestv[s][v] = -3.402823e38f; besti[s][v] = 0; }

    for (int nb = 0; nb < K_ / BN; ++nb) {
        const int n0 = nb * BN;

        v8f acc[4];
#pragma unroll
        for (int j = 0; j < 4; ++j) acc[j] = (v8f)0.0f;

        // pipeline prologue: chunk 0 -> buffer 0
        stageA(0, 0);
        stageB(0, n0, 0);
        asm volatile("s_wait_asynccnt 0" ::: "memory");
        __syncthreads();

        for (int kci = 0; kci < KCH; ++kci) {
            const int cur = kci & 1;
            if (kci + 1 < KCH) {            // prefetch next chunk into other buffer
                stageA(1 - cur, (kci + 1) * BK);
                stageB(1 - cur, n0, (kci + 1) * BK);
            }

            // WMMA over current chunk. A frag: (M=l16, K=kk+2*lhi+{0,1});
            // B frag: (N=l16, K=kk+2*lhi+{0,1}).  Both ds_load_b64.
#pragma unroll
            for (int kk = 0; kk < BK; kk += 4) {
                const v2f aF = *(const v2f*)&Asub[cur][wm * 16 + l16][kk + lhi * 2];
                v2f bF[4];
#pragma unroll
                for (int j = 0; j < 4; ++j)
                    bF[j] = *(const v2f*)&Bsub[cur][wn * 64 + j * 16 + l16][kk + lhi * 2];
#pragma unroll
                for (int j = 0; j < 4; ++j)
                    acc[j] = __builtin_amdgcn_wmma_f32_16x16x4_f32(
                        false, aF, false, bF[j], (short)0, acc[j], false, false);
            }

            asm volatile("s_wait_asynccnt 0" ::: "memory");  // next buffer landed
            __syncthreads();                                  // all waves done with cur
        }

        // gumbel + running argmax.  C/D layout: VGPR v, lane l -> M = v+8*(l>=16),
        // N = l&15.  n ascends (j, nb) so strict '>' keeps first max (jnp.argmax).
        const int nbase = n0 + wn * 64 + l16;
#pragma unroll
        for (int s = 0; s < NS_; ++s) {
#pragma unroll
            for (int v = 0; v < 8; ++v) {
                const int r = row0 + wm * 16 + v + (lhi << 3);
                const size_t gbase = ((size_t)(r * NS_ + s)) * K_ + (size_t)nbase;
#pragma unroll
                for (int j = 0; j < 4; ++j) {
                    const float g = acc[j][v] +
                        __builtin_nontemporal_load(&gn[gbase + j * 16]);
                    if (g > bestv[s][v]) { bestv[s][v] = g; besti[s][v] = nbase + j * 16; }
                }
            }
        }
    }

    // cross-lane argmax reduction within each 16-lane group (same M row)
#pragma unroll
    for (int s = 0; s < NS_; ++s) {
#pragma unroll
        for (int v = 0; v < 8; ++v) {
            float bv = bestv[s][v];
            int   bi = besti[s][v];
#pragma unroll
            for (int off = 8; off >= 1; off >>= 1) {
                const float ov = __shfl_xor(bv, off, 32);
                const int   oi = __shfl_xor(bi, off, 32);
                if (ov > bv || (ov == bv && oi < bi)) { bv = ov; bi = oi; }
            }
            if (l16 == 0) {
                const int r = row0 + wm * 16 + v + (lhi << 3);
                idx_out[r * NS_ + s] = bi;
            }
        }
    }
}

// zero-fill the one-hot output region (134.2M floats), non-temporal wide stores
__global__ void zero_indices_kernel(v4f* __restrict__ p, long n4) {
    long i = (long)blockIdx.x * blockDim.x + threadIdx.x;
    const long stride = (long)gridDim.x * blockDim.x;
    const v4f z = (v4f)0.0f;
    for (; i < n4; i += stride) __builtin_nontemporal_store(z, &p[i]);
}

// scatter 1.0 at argmax and gather the embedding row (768 floats = 192 float4)
__global__ __launch_bounds__(192)
void scatter_gather_kernel(const int* __restrict__ idx_ws,
                           const float* __restrict__ emb_tab,
                           float* __restrict__ out_emb,
                           float* __restrict__ out_ind) {
    const int p   = blockIdx.x;           // r*NS + s
    const int idx = idx_ws[p];
    const int t   = threadIdx.x;          // 0..191
    const v4f v = *(const v4f*)&emb_tab[(size_t)idx * D_ + t * 4];   // L2-resident table
    __builtin_nontemporal_store(v, (v4f*)&out_emb[(size_t)p * D_ + t * 4]);
    if (t == 0) __builtin_nontemporal_store(1.0f, &out_ind[(size_t)p * K_ + idx]);
}

extern "C" void kernel_launch(void* const* d_in, const int* in_sizes, int n_in,
                              void* d_out, int out_size, void* d_ws, size_t ws_size,
                              hipStream_t stream) {
    const float* x  = (const float*)d_in[0];   // [B,S,D]
    const float* w  = (const float*)d_in[1];   // [K,D]
    const float* em = (const float*)d_in[2];   // [K,D]
    const float* gn = (const float*)d_in[3];   // [B,S,NS,K]

    float* out      = (float*)d_out;
    float* out_emb  = out;                                   // [B,S,NS,D]
    float* out_ind  = out + (size_t)M_ * NS_ * D_;           // [B,S,NS,K]
    int*   idx_ws   = (int*)d_ws;                            // [M, NS] = 64 KB

    gumbel_gemm_argmax_kernel<<<M_ / BM, 256, 0, stream>>>(x, w, gn, idx_ws);
    zero_indices_kernel<<<4096, 256, 0, stream>>>((v4f*)out_ind,
                                                  (long)M_ * NS_ * K_ / 4);
    scatter_gather_kernel<<<M_ * NS_, 192, 0, stream>>>(idx_ws, em, out_emb, out_ind);
}